// MultiHead_HGTLayer_simplified_32547262169452
// MI455X (gfx1250) — compile-verified
//
#include <hip/hip_runtime.h>
#include <hip/hip_bf16.h>
#include <math.h>

// Problem constants (fixed by the reference)
#define NN    40000
#define EE    640000
#define IN_F  128
#define OUT_F 128
#define NT    8
#define NR    16
#define NH    8
#define DKH   16   // DK = OUT/H

typedef __attribute__((ext_vector_type(2))) float v2f;
typedef __attribute__((ext_vector_type(8))) float v8f;

// ---------------------------------------------------------------------------
// helpers
// ---------------------------------------------------------------------------
__device__ inline void atomicMaxFloat(float* addr, float val) {
    int* iaddr = reinterpret_cast<int*>(addr);
    int assumed = *iaddr;
    while (val > __int_as_float(assumed)) {
        int old = atomicCAS(iaddr, assumed, __float_as_int(val));
        if (old == assumed) break;
        assumed = old;
    }
}

// ---------------------------------------------------------------------------
// K0: init segment-softmax state + mailbox
// ---------------------------------------------------------------------------
__global__ void hgt_init_kernel(float* __restrict__ amax,
                                float* __restrict__ denom,
                                float* __restrict__ tmail) {
    int i = blockIdx.x * blockDim.x + threadIdx.x;
    if (i < NN * NH) {
        amax[i]  = -INFINITY;
        denom[i] = 0.0f;
    }
    if (i < NN * OUT_F) tmail[i] = 0.0f;
}

// ---------------------------------------------------------------------------
// K1: type-indexed k/q/v projections via type-masked fp32 WMMA.
//     blockIdx.x = node tile (16 nodes), blockIdx.y = {0:k, 1:q, 2:v}
//     8 waves per block, wave w owns output columns [16w, 16w+16).
//     D[tile] = sum_t mask_{type==t}(X[tile]) @ W[t]   (single accumulator)
// ---------------------------------------------------------------------------
__global__ void hgt_proj_kernel(const float* __restrict__ x,
                                const float* __restrict__ Wk,
                                const float* __restrict__ Wq,
                                const float* __restrict__ Wv,
                                const int*   __restrict__ ntype,
                                float* __restrict__ kn,
                                float* __restrict__ qn,
                                float* __restrict__ vn) {
    __shared__ float xs[16 * 132];   // padded row stride 132 -> conflict-free A loads
    __shared__ int   ts[16];
    __shared__ unsigned tmask_s;

    const int node0 = blockIdx.x * 16;
    const float* W   = (blockIdx.y == 0) ? Wk : (blockIdx.y == 1) ? Wq : Wv;
    float*       dst = (blockIdx.y == 0) ? kn : (blockIdx.y == 1) ? qn : vn;

    const int tid = threadIdx.x;
    for (int i = tid; i < 16 * IN_F; i += 256) {
        int r = i >> 7, c = i & 127;
        xs[r * 132 + c] = x[(node0 + r) * IN_F + c];
    }
    if (tid < 16) ts[tid] = ntype[node0 + tid];
    __syncthreads();
    if (tid == 0) {
        unsigned m = 0;
        for (int i = 0; i < 16; ++i) m |= 1u << ts[i];
        tmask_s = m;
    }
    __syncthreads();
    const unsigned tmask = tmask_s;

    const int lane = tid & 31;
    const int wave = tid >> 5;        // column tile
    const int col0 = wave * 16;
    const int m    = lane & 15;       // A: M = lane&15 ; B: N = lane&15
    const int kh   = lane >> 4;       // A/B K-half select (K = 2*kh, 2*kh+1)

    v8f acc = {};
    for (int t = 0; t < NT; ++t) {
        if (!((tmask >> t) & 1u)) continue;
        const float* Wt = W + (size_t)t * IN_F * OUT_F;
        const float rmask = (ts[m] == t) ? 1.0f : 0.0f;
        #pragma unroll 4
        for (int kb = 0; kb < IN_F; kb += 4) {
            v2f a, b;
            a.x = xs[m * 132 + kb + 2 * kh + 0] * rmask;
            a.y = xs[m * 132 + kb + 2 * kh + 1] * rmask;
            b.x = Wt[(kb + 2 * kh + 0) * OUT_F + col0 + m];
            b.y = Wt[(kb + 2 * kh + 1) * OUT_F + col0 + m];
            acc = __builtin_amdgcn_wmma_f32_16x16x4_f32(
                false, a, false, b, (short)0, acc, false, false);
        }
    }
    // D layout: VGPR i -> M=i (lanes 0-15), M=i+8 (lanes 16-31); N = lane&15
    const int rowoff = (lane >> 4) * 8;
    #pragma unroll
    for (int i = 0; i < 8; ++i)
        dst[(size_t)(node0 + rowoff + i) * OUT_F + col0 + m] = acc[i];
}

// ---------------------------------------------------------------------------
// K2: per (edge, head) attention logit  attn = k . (Att[r,h] q) * pri / sqrt(OUT)
//     relation_att staged in LDS swizzled [(d*16+c)*128 + rh] (rh fastest ->
//     32 lanes with distinct (r,h) hit distinct banks). Atomic segment max.
// ---------------------------------------------------------------------------
__global__ void hgt_attn_kernel(const float* __restrict__ kn,
                                const float* __restrict__ qn,
                                const float* __restrict__ rel_att,
                                const float* __restrict__ rel_pri,
                                const int*   __restrict__ src,
                                const int*   __restrict__ dstn,
                                const int*   __restrict__ etype,
                                float* __restrict__ attn,
                                float* __restrict__ amax) {
    extern __shared__ float sm[];
    float* As  = sm;                   // 256*128 floats, swizzled
    float* pri = sm + 256 * 128;       // 128 floats
    for (int i = threadIdx.x; i < NR * NH * DKH * DKH; i += blockDim.x) {
        int rh = i >> 8, dc = i & 255;
        As[dc * 128 + rh] = rel_att[i];
    }
    for (int i = threadIdx.x; i < NR * NH; i += blockDim.x) pri[i] = rel_pri[i];
    __syncthreads();

    const float inv_sqrt = 0.08838834764831845f;  // 1/sqrt(128)
    const int total = EE * NH;
    for (int idx = blockIdx.x * blockDim.x + threadIdx.x; idx < total;
         idx += gridDim.x * blockDim.x) {
        const int e = idx >> 3, h = idx & 7;
        const int r  = etype[e];
        const int rh = r * NH + h;
        const int s  = src[e], d = dstn[e];
        const float* krow = kn + (size_t)s * OUT_F + h * DKH;
        const float* qrow = qn + (size_t)d * OUT_F + h * DKH;
        float q[16], k[16];
        #pragma unroll
        for (int c = 0; c < 16; ++c) { q[c] = qrow[c]; k[c] = krow[c]; }
        float acc = 0.0f;
        #pragma unroll
        for (int dd = 0; dd < 16; ++dd) {
            float w = 0.0f;
            #pragma unroll
            for (int c = 0; c < 16; ++c)
                w = fmaf(As[(dd * 16 + c) * 128 + rh], q[c], w);
            acc = fmaf(k[dd], w, acc);
        }
        const float a = acc * pri[rh] * inv_sqrt;
        attn[idx] = a;
        atomicMaxFloat(&amax[d * NH + h], a);
    }
}

// ---------------------------------------------------------------------------
// K3: p = exp(attn - amax[dst]) ; denom[dst] += p  (in-place overwrite attn)
// ---------------------------------------------------------------------------
__global__ void hgt_expsum_kernel(float* __restrict__ attn,
                                  const float* __restrict__ amax,
                                  const int* __restrict__ dstn,
                                  float* __restrict__ denom) {
    const int total = EE * NH;
    for (int idx = blockIdx.x * blockDim.x + threadIdx.x; idx < total;
         idx += gridDim.x * blockDim.x) {
        const int e = idx >> 3, h = idx & 7;
        const int d = dstn[e];
        const float p = __expf(attn[idx] - amax[d * NH + h]);
        attn[idx] = p;
        atomicAdd(&denom[d * NH + h], p);
    }
}

// ---------------------------------------------------------------------------
// K4: message: v2 = v_row @ Msg[r,h]; alpha = p/denom; mailbox += alpha*v2
// ---------------------------------------------------------------------------
__global__ void hgt_msg_kernel(const float* __restrict__ vn,
                               const float* __restrict__ rel_msg,
                               const int*   __restrict__ src,
                               const int*   __restrict__ dstn,
                               const int*   __restrict__ etype,
                               const float* __restrict__ p,
                               const float* __restrict__ denom,
                               float* __restrict__ tmail) {
    extern __shared__ float sm[];
    float* Ms = sm;  // 256*128 floats, swizzled like K2
    for (int i = threadIdx.x; i < NR * NH * DKH * DKH; i += blockDim.x) {
        int rh = i >> 8, dc = i & 255;
        Ms[dc * 128 + rh] = rel_msg[i];
    }
    __syncthreads();

    const int total = EE * NH;
    for (int idx = blockIdx.x * blockDim.x + threadIdx.x; idx < total;
         idx += gridDim.x * blockDim.x) {
        const int e = idx >> 3, h = idx & 7;
        const int r  = etype[e];
        const int rh = r * NH + h;
        const int s  = src[e], d = dstn[e];
        const float alpha = p[idx] / denom[d * NH + h];
        const float* vrow = vn + (size_t)s * OUT_F + h * DKH;
        float v[16];
        #pragma unroll
        for (int dd = 0; dd < 16; ++dd) v[dd] = vrow[dd];
        float* trow = tmail + (size_t)d * OUT_F + h * DKH;
        #pragma unroll
        for (int c = 0; c < 16; ++c) {
            float m = 0.0f;
            #pragma unroll
            for (int dd = 0; dd < 16; ++dd)
                m = fmaf(v[dd], Ms[(dd * 16 + c) * 128 + rh], m);
            atomicAdd(&trow[c], alpha * m);
        }
    }
}

// ---------------------------------------------------------------------------
// K5: output projection (type-masked WMMA) + per-(row,head) sigmoid skip gate.
//     Column tile == head (DK == 16), so gate is applied per D element.
// ---------------------------------------------------------------------------
__global__ void hgt_outproj_kernel(const float* __restrict__ tmail,
                                   const float* __restrict__ Wa,
                                   const float* __restrict__ skip,
                                   const int*   __restrict__ ntype,
                                   float* __restrict__ out) {
    __shared__ float xs[16 * 132];
    __shared__ int   ts[16];
    __shared__ unsigned tmask_s;

    const int node0 = blockIdx.x * 16;
    const int tid = threadIdx.x;
    for (int i = tid; i < 16 * OUT_F; i += 256) {
        int r = i >> 7, c = i & 127;
        xs[r * 132 + c] = tmail[(size_t)(node0 + r) * OUT_F + c];
    }
    if (tid < 16) ts[tid] = ntype[node0 + tid];
    __syncthreads();
    if (tid == 0) {
        unsigned m = 0;
        for (int i = 0; i < 16; ++i) m |= 1u << ts[i];
        tmask_s = m;
    }
    __syncthreads();
    const unsigned tmask = tmask_s;

    const int lane = tid & 31;
    const int wave = tid >> 5;       // column tile == head index
    const int col0 = wave * 16;
    const int m    = lane & 15;
    const int kh   = lane >> 4;

    v8f acc = {};
    for (int t = 0; t < NT; ++t) {
        if (!((tmask >> t) & 1u)) continue;
        const float* Wt = Wa + (size_t)t * OUT_F * OUT_F;
        const float rmask = (ts[m] == t) ? 1.0f : 0.0f;
        #pragma unroll 4
        for (int kb = 0; kb < OUT_F; kb += 4) {
            v2f a, b;
            a.x = xs[m * 132 + kb + 2 * kh + 0] * rmask;
            a.y = xs[m * 132 + kb + 2 * kh + 1] * rmask;
            b.x = Wt[(kb + 2 * kh + 0) * OUT_F + col0 + m];
            b.y = Wt[(kb + 2 * kh + 1) * OUT_F + col0 + m];
            acc = __builtin_amdgcn_wmma_f32_16x16x4_f32(
                false, a, false, b, (short)0, acc, false, false);
        }
    }
    const int rowoff = (lane >> 4) * 8;
    #pragma unroll
    for (int i = 0; i < 8; ++i) {
        const int rlocal = rowoff + i;
        const float g = 1.0f / (1.0f + __expf(-skip[ts[rlocal] * NH + wave]));
        out[(size_t)(node0 + rlocal) * OUT_F + col0 + m] = acc[i] * g;
    }
}

// ---------------------------------------------------------------------------
// launch
// ---------------------------------------------------------------------------
extern "C" void kernel_launch(void* const* d_in, const int* in_sizes, int n_in,
                              void* d_out, int out_size, void* d_ws, size_t ws_size,
                              hipStream_t stream) {
    const float* h     = (const float*)d_in[0];
    const float* kw    = (const float*)d_in[1];
    const float* qw    = (const float*)d_in[2];
    const float* vw    = (const float*)d_in[3];
    const float* aw    = (const float*)d_in[4];
    const float* pri   = (const float*)d_in[5];
    const float* ratt  = (const float*)d_in[6];
    const float* rmsg  = (const float*)d_in[7];
    const float* skip  = (const float*)d_in[8];
    const int*   src   = (const int*)d_in[9];
    const int*   dstn  = (const int*)d_in[10];
    const int*   ntype = (const int*)d_in[11];
    const int*   etype = (const int*)d_in[12];
    float* out = (float*)d_out;

    float* ws    = (float*)d_ws;
    float* kn    = ws;
    float* qn    = kn + (size_t)NN * OUT_F;
    float* vn    = qn + (size_t)NN * OUT_F;
    float* attn  = vn + (size_t)NN * OUT_F;     // E*H, later holds p
    float* amax  = attn + (size_t)EE * NH;
    float* denom = amax + (size_t)NN * NH;
    float* tmail = denom + (size_t)NN * NH;

    const int initN = NN * OUT_F;
    hgt_init_kernel<<<(initN + 255) / 256, 256, 0, stream>>>(amax, denom, tmail);

    dim3 gproj(NN / 16, 3);
    hgt_proj_kernel<<<gproj, 256, 0, stream>>>(h, kw, qw, vw, ntype, kn, qn, vn);

    const size_t relsm = (size_t)(256 * 128 + 128) * sizeof(float);  // 128.5 KB LDS
    hgt_attn_kernel<<<512, 256, relsm, stream>>>(kn, qn, ratt, pri, src, dstn,
                                                 etype, attn, amax);
    hgt_expsum_kernel<<<512, 256, 0, stream>>>(attn, amax, dstn, denom);
    hgt_msg_kernel<<<512, 256, relsm, stream>>>(vn, rmsg, src, dstn, etype,
                                                attn, denom, tmail);
    hgt_outproj_kernel<<<NN / 16, 256, 0, stream>>>(tmail, aw, skip, ntype, out);
}